// AttentionHead_13065290514585
// MI455X (gfx1250) — compile-verified
//
#include <hip/hip_runtime.h>
#include <stdint.h>

typedef __attribute__((ext_vector_type(2))) float        v2f;
typedef __attribute__((ext_vector_type(8))) float        v8f;
typedef __attribute__((ext_vector_type(4))) unsigned int v4u;
typedef __attribute__((ext_vector_type(4))) int          v4i;
typedef __attribute__((ext_vector_type(8))) int          v8i;

#define CIN   512
#define COUT  64
#define BSZ   8
#define HDIM  28
#define HW    784            // 28*28
#define REP   1024
#define KCONV 4608           // CIN*9

// ---------------------------------------------------------------------------
// TDM: issue an async 2D tile load (16 rows x 128 floats) of a row-major
// (1024, K) f32 matrix into LDS. D# bit layout per CDNA5 ISA §8.3/8.4.
// ---------------------------------------------------------------------------
static __device__ __forceinline__ void
tdm_load_tile(const float* gsrc, unsigned lds_off, int K) {
    unsigned long long ga = (unsigned long long)(uintptr_t)gsrc;
    v4u g0;
    g0[0] = 1u;                                               // count=1, user desc
    g0[1] = lds_off;                                          // lds_addr (bytes)
    g0[2] = (unsigned)ga;                                     // global_addr[31:0]
    g0[3] = (unsigned)((ga >> 32) & 0x01FFFFFFu) | (2u << 30);// addr[56:32] | type=2
    v8i g1;
    g1[0] = (int)(2u << 16);                                  // data_size = 4 bytes
    g1[1] = (int)(((unsigned)K & 0xFFFFu) << 16);             // tensor_dim0 lo16
    g1[2] = (int)(((unsigned)K >> 16) | (1024u << 16));       // dim0 hi16 | tensor_dim1 lo16
    g1[3] = (int)(128u << 16);                                // tile_dim0 = 128
    g1[4] = 16;                                               // tile_dim1 = 16
    g1[5] = K;                                                // tensor_dim0_stride lo32
    g1[6] = 0;
    g1[7] = 0;
    v4i z4 = {0, 0, 0, 0};
#if defined(__clang_major__) && (__clang_major__ >= 23)
    v8i z8 = {0, 0, 0, 0, 0, 0, 0, 0};
    __builtin_amdgcn_tensor_load_to_lds(g0, g1, z4, z4, z8, 0);
#else
    __builtin_amdgcn_tensor_load_to_lds(g0, g1, z4, z4, 0);
#endif
}

// ---------------------------------------------------------------------------
// 1) Transpose conv weights (COUT, CIN*9) -> (CIN*9, COUT).
// ---------------------------------------------------------------------------
__global__ void __launch_bounds__(256)
transpose_convw(const float* __restrict__ cw, float* __restrict__ wT) {
    int idx = blockIdx.x * 256 + threadIdx.x;   // 294912 = 4608*64
    int k = idx >> 6;
    int n = idx & 63;
    wT[idx] = cw[n * KCONV + k];                // wT[k*64 + n]
}

// ---------------------------------------------------------------------------
// 2) 3x3 SAME conv as implicit GEMM with V_WMMA_F32_16X16X4_F32.
//    784 = 16*49 -> a 16-row M-tile never crosses a batch boundary, so the
//    batch index is BLOCK-UNIFORM: scalar (SGPR) base pointers + 32-bit lane
//    offsets -> saddr-form global loads, no 64-bit per-lane address math.
//    K super-step of 36 = lcm(4,9): per-lane im2col tap table is loop
//    invariant; borders handled branchlessly by 0/1 mask multiply.
// ---------------------------------------------------------------------------
__global__ void __launch_bounds__(128)
conv3x3_wmma(const float* __restrict__ x, const float* __restrict__ wT,
             const float* __restrict__ bias, float* __restrict__ am) {
    const int mtile = blockIdx.x;             // 0..391
    const int lane  = threadIdx.x & 31;
    const int ntile = threadIdx.x >> 5;       // 0..3
    const int l15   = lane & 15;
    const int khalf = lane >> 4;              // A: lanes16-31 hold K+2,K+3

    const int b     = mtile / 49;             // block-uniform batch index
    const int hwrow = (mtile - b * 49) * 16 + l15;   // 0..783 within image
    const int hh    = hwrow / HDIM;
    const int ww    = hwrow - hh * HDIM;
    const float* __restrict__ xb = x + (size_t)b * CIN * HW;  // scalar base
    const int n = ntile * 16 + l15;           // cout column for B loads

    // Per-lane im2col table for super-step offset kk = khalf*2 + j + 4u (0..35).
    int   offt[18];
    float mskt[18];
#pragma unroll
    for (int u = 0; u < 9; ++u) {
#pragma unroll
        for (int j = 0; j < 2; ++j) {
            int kk  = khalf * 2 + j + 4 * u;  // 0..35
            int cio = kk / 9;                 // 0..3
            int tap = kk - cio * 9;
            int dh  = tap / 3;
            int dw  = tap - dh * 3;
            int ih  = hh + dh - 1;
            int iw  = ww + dw - 1;
            bool v  = ((unsigned)ih < 28u) && ((unsigned)iw < 28u);
            offt[u * 2 + j] = cio * HW + (v ? (ih * HDIM + iw) : 0);
            mskt[u * 2 + j] = v ? 1.0f : 0.0f;
        }
    }

    v8f acc = {};
    for (int kb = 0; kb < KCONV; kb += 36) {           // 128 iterations
        const int cibase = (kb / 9) * HW;              // scalar (kb%9 == 0)
        const float* __restrict__ wq = wT + (size_t)kb * COUT;  // scalar base
#pragma unroll
        for (int u = 0; u < 9; ++u) {
            v2f a;
            a[0] = xb[cibase + offt[u * 2 + 0]] * mskt[u * 2 + 0];
            a[1] = xb[cibase + offt[u * 2 + 1]] * mskt[u * 2 + 1];
            v2f bv;
            bv[0] = wq[(4 * u + khalf) * COUT + n];
            bv[1] = wq[(4 * u + 2 + khalf) * COUT + n];
            acc = __builtin_amdgcn_wmma_f32_16x16x4_f32(false, a, false, bv,
                                                        (short)0, acc, false, false);
        }
    }

    // D layout: VGPR r -> M = r (+8 for upper lanes), N = lane%16
    const int   co  = n;
    const float bb  = bias[co];
    const int   mh0 = (mtile - b * 49) * 16 + khalf * 8;      // local row base
    float* __restrict__ amb = am + (size_t)b * COUT * HW;     // scalar base
#pragma unroll
    for (int r = 0; r < 8; ++r)
        amb[co * HW + mh0 + r] = acc[r] + bb;
}

// ---------------------------------------------------------------------------
// 3) Fused outer-product + global max pool:
//    xpool[b,cin,cout] = max_hw am[b,cout,hw] * x[b,cin,hw]
// ---------------------------------------------------------------------------
__global__ void __launch_bounds__(256)
pool_outer_max(const float* __restrict__ x, const float* __restrict__ am,
               float* __restrict__ xpool) {
    const int cin = blockIdx.x;               // 512
    const int b   = blockIdx.y;               // 8
    __shared__ float xs[HW];
    __shared__ float red[256];
    const int t = threadIdx.x;

    const float* __restrict__ xr = x + ((size_t)b * CIN + cin) * HW;
    for (int i = t; i < HW; i += 256) xs[i] = xr[i];
    __syncthreads();

    const int co = t >> 2;                    // 0..63
    const int s  = t & 3;                     // 0..3
    const float* __restrict__ ar = am + ((size_t)b * COUT + co) * HW;
    float mx = -3.402823466e38f;
    for (int hw = s; hw < HW; hw += 4)
        mx = fmaxf(mx, ar[hw] * xs[hw]);
    red[t] = mx;
    __syncthreads();
    if (s == 0) {
        float m0 = fmaxf(fmaxf(red[t], red[t + 1]), fmaxf(red[t + 2], red[t + 3]));
        xpool[((size_t)b * CIN + cin) * COUT + co] = m0;
    }
}

// ---------------------------------------------------------------------------
// 4) Split-K WMMA FC partial GEMM with TDM double-buffered weight streaming:
//    out[m,n] = sum_k in[m,k] * w[n,k]   (m<8, n<1024)
//    A = 16 weight rows (TDM tile in LDS), B = activations (m padded to 16).
//    Scalar base + 32-bit lane offsets for B loads and stores.
// ---------------------------------------------------------------------------
__global__ void __launch_bounds__(32)
fc_wmma_partial(const float* __restrict__ in,   // (8, K) row-major
                const float* __restrict__ w,    // (1024, K) row-major
                float* __restrict__ partial,    // (Kchunks, 8, 1024)
                int K, int Kc) {
    __shared__ float lds[2][16 * 128];          // 2 x 8 KB tiles
    const int ntile = blockIdx.x;               // 0..63
    const int kc    = blockIdx.y;               // split-K chunk
    const int lane  = threadIdx.x;
    const int l15   = lane & 15;
    const int khalf = lane >> 4;
    const int k0    = kc * Kc;
    const int nch   = Kc >> 7;                  // chunks of 128 K

    const float* wtile0 = w + (size_t)(ntile * 16) * K;

    // prime the pipeline
    tdm_load_tile(wtile0 + k0, (unsigned)(uintptr_t)&lds[0][0], K);

    const int   m   = l15;                      // B column = batch (valid m<8)
    const float msk = (m < 8) ? 1.0f : 0.0f;
    const int   mK  = (m < 8 ? m : 0) * K;      // 32-bit lane offset base

    v8f acc = {};
    for (int c = 0; c < nch; ++c) {
        const int kb = k0 + (c << 7);
        if (c + 1 < nch) {
            tdm_load_tile(wtile0 + kb + 128,
                          (unsigned)(uintptr_t)&lds[(c + 1) & 1][0], K);
            __builtin_amdgcn_s_wait_tensorcnt(1);   // current tile arrived
        } else {
            __builtin_amdgcn_s_wait_tensorcnt(0);
        }
        __asm__ volatile("" ::: "memory");

        const float* __restrict__ arow = &lds[c & 1][0] + l15 * 128 + khalf * 2;
        const int boff = mK + kb + khalf;           // lane offset into `in`
#pragma unroll 4
        for (int kk = 0; kk < 128; kk += 4) {
            v2f a;
            a[0] = arow[kk];
            a[1] = arow[kk + 1];
            v2f bv;
            bv[0] = in[boff + kk]     * msk;
            bv[1] = in[boff + kk + 2] * msk;
            acc = __builtin_amdgcn_wmma_f32_16x16x4_f32(false, a, false, bv,
                                                        (short)0, acc, false, false);
        }
        // LDS reads must land before TDM overwrites this buffer next iter
        __asm__ volatile("s_wait_dscnt 0" ::: "memory");
    }

    // D[M=n_local][N=m]: lane holds m = l15, n_local = r + 8*khalf
    float* __restrict__ pbase = partial + (size_t)kc * 8 * REP;   // scalar base
    if (m < 8) {
        const int nn0 = ntile * 16 + khalf * 8;
#pragma unroll
        for (int r = 0; r < 8; ++r)
            pbase[m * REP + nn0 + r] = acc[r];
    }
}

// ---------------------------------------------------------------------------
// 5) Split-K reduction + bias + ReLU.
// ---------------------------------------------------------------------------
__global__ void __launch_bounds__(256)
fc_reduce_bias_relu(const float* __restrict__ partial,
                    const float* __restrict__ bias,
                    float* __restrict__ out, int Kchunks) {
    int idx = blockIdx.x * 256 + threadIdx.x;   // 8192 = 8*1024
    int n = idx & 1023;
    float s = bias[n];
    int m = idx >> 10;
    for (int kc = 0; kc < Kchunks; ++kc)
        s += partial[((size_t)kc * 8 + m) * REP + n];
    out[idx] = s > 0.0f ? s : 0.0f;
}

// ---------------------------------------------------------------------------
extern "C" void kernel_launch(void* const* d_in, const int* in_sizes, int n_in,
                              void* d_out, int out_size, void* d_ws, size_t ws_size,
                              hipStream_t stream) {
    (void)in_sizes; (void)n_in; (void)out_size; (void)ws_size;

    const float* x      = (const float*)d_in[0];   // (8,512,28,28)
    const float* conv_w = (const float*)d_in[1];   // (64,512,3,3)
    const float* conv_b = (const float*)d_in[2];   // (64,)
    const float* fc6_w  = (const float*)d_in[3];   // (1024,32768)
    const float* fc6_b  = (const float*)d_in[4];   // (1024,)
    const float* fc7_w  = (const float*)d_in[5];   // (1024,1024)
    const float* fc7_b  = (const float*)d_in[6];   // (1024,)

    float* out   = (float*)d_out;                  // (8,1024)  first 8192
    float* xpool = out + BSZ * REP;                // (8,512,64) == flat (8,32768)

    float* wT      = (float*)d_ws;                 // 294912 floats (1.18 MB)
    float* am      = wT + KCONV * COUT;            // 401408 floats (1.6 MB)
    float* partial = am + (size_t)BSZ * COUT * HW; // up to 131072 floats (512 KB)
    float* h       = partial + 16 * 8 * REP;       // 8192 floats (32 KB)

    // conv weight transpose -> conv -> fused outer*max pool
    transpose_convw<<<KCONV * COUT / 256, 256, 0, stream>>>(conv_w, wT);
    conv3x3_wmma<<<392, 128, 0, stream>>>(x, wT, conv_b, am);
    pool_outer_max<<<dim3(CIN, BSZ), 256, 0, stream>>>(x, am, xpool);

    // fc6: K=32768, split-K 16 (1024 waves, TDM-streamed 134 MB weight matrix)
    fc_wmma_partial<<<dim3(REP / 16, 16), 32, 0, stream>>>(xpool, fc6_w, partial,
                                                           CIN * COUT, CIN * COUT / 16);
    fc_reduce_bias_relu<<<BSZ * REP / 256, 256, 0, stream>>>(partial, fc6_b, h, 16);

    // fc7: K=1024, split-K 4
    fc_wmma_partial<<<dim3(REP / 16, 4), 32, 0, stream>>>(h, fc7_w, partial,
                                                          REP, REP / 4);
    fc_reduce_bias_relu<<<BSZ * REP / 256, 256, 0, stream>>>(partial, fc7_b, out, 4);
}